// GCNModule_34677565948514
// MI455X (gfx1250) — compile-verified
//
#include <hip/hip_runtime.h>

#define DF 128     // feature dim (fixed by reference)
#define SLABS 8    // 16-row slabs per block -> 128 rows/block

typedef float v2f __attribute__((ext_vector_type(2)));
typedef float v8f __attribute__((ext_vector_type(8)));

// ---------------------------------------------------------------- utilities
__global__ void zero_f32(float* __restrict__ p, int n) {
    int i = blockIdx.x * blockDim.x + threadIdx.x;
    if (i < n) p[i] = 0.0f;
}

// degree over destinations (float atomics exact for deg << 2^24)
__global__ void deg_count(const int* __restrict__ dst, float* __restrict__ deg, int E) {
    int e = blockIdx.x * blockDim.x + threadIdx.x;
    if (e < E) unsafeAtomicAdd(&deg[dst[e]], 1.0f);
}

// dinv = rsqrt(deg + 1)   (+1 = self loop; deg >= 1 so no zero guard needed)
__global__ void deg_finalize(float* __restrict__ dinv, int n) {
    int i = blockIdx.x * blockDim.x + threadIdx.x;
    if (i < n) dinv[i] = rsqrtf(dinv[i] + 1.0f);
}

// --------------------------------------------------------- async slab copy
// Copy one 16xDF f32 slab (8 KB) into LDS with GLOBAL_LOAD_ASYNC_TO_LDS_B128.
// 256 threads x 16 B x 2 chunks. OOB rows are clamp-addressed (EXEC stays
// all-ones, per-wave ASYNCcnt deterministic); their results are discarded at
// the guarded store. Whole-tensor byte offsets fit in u32 (N*DF*4 = 51.2 MB).
__device__ __forceinline__ void async_slab_copy(const float* __restrict__ A,
                                                float* lds_dst, int rowbase,
                                                int nrows, int tid) {
#pragma unroll
    for (int c = 0; c < 2; ++c) {
        int elt = (c * 256 + tid) * 4;           // float index within slab
        int r   = rowbase + (elt >> 7);          // DF floats per row
        int rc  = (r < nrows) ? r : (nrows - 1); // clamp, keep EXEC all-ones
        unsigned lds  = (unsigned)(uintptr_t)(lds_dst + elt);
        unsigned goff = (unsigned)(((size_t)rc * DF + (elt & (DF - 1))) * 4u);
        asm volatile("global_load_async_to_lds_b128 %0, %1, %2"
                     :: "v"(lds), "v"(goff), "s"(A) : "memory");
    }
}

// ------------------------------------------------------- GEMM: Y = A @ W
// Block = SLABS x 16 rows. 8 waves -> 8 column tiles of 16. B fragments for
// the full K=128 live in 64 VGPRs per lane, loaded once and reused across all
// slabs. A slabs stream through double-buffered LDS via async-to-LDS copies.
__global__ __launch_bounds__(256) void gemm_wmma_f32(
    const float* __restrict__ A, const float* __restrict__ W,
    float* __restrict__ Y, int nrows) {
    __shared__ float As[2][16 * DF];  // double buffer, 2 x 8 KB

    const int tid     = threadIdx.x;
    const int wave    = tid >> 5;
    const int lane    = tid & 31;
    const int colbase = wave << 4;
    const int m       = lane & 15;         // row within tile (A) / col (B,C)
    const int khalf   = (lane >> 4) << 1;  // lanes 0-15: K+{0,1}; 16-31: K+{2,3}

    // preload all B fragments (128x16 column slice) into registers
    v2f bf[DF / 4];
#pragma unroll
    for (int kb = 0; kb < DF / 4; ++kb) {
        const int k0 = kb * 4 + khalf;
        bf[kb][0] = W[(size_t)k0 * DF + colbase + m];
        bf[kb][1] = W[(size_t)(k0 + 1) * DF + colbase + m];
    }

    const int blockrow = blockIdx.x * (16 * SLABS);

    // prologue: start slab 0 copy
    async_slab_copy(A, As[0], blockrow, nrows, tid);

#pragma unroll 1
    for (int s = 0; s < SLABS; ++s) {
        const int buf     = s & 1;
        const int rowbase = blockrow + s * 16;

        if (s + 1 < SLABS) {
            async_slab_copy(A, As[buf ^ 1], rowbase + 16, nrows, tid);
            asm volatile("s_wait_asynccnt 0x2" ::: "memory");  // slab s landed
        } else {
            asm volatile("s_wait_asynccnt 0x0" ::: "memory");
        }
        __syncthreads();  // slab s visible from all waves

        v8f c = {};  // 16x16 f32 accumulator (8 VGPRs)
#pragma unroll
        for (int kb = 0; kb < DF / 4; ++kb) {
            const int k0 = kb * 4 + khalf;
            v2f a;
            // A fragment (16x4, ISA layout): lane m holds (M=m, K=k0..k0+1)
            a[0] = As[buf][m * DF + k0];
            a[1] = As[buf][m * DF + k0 + 1];
            c = __builtin_amdgcn_wmma_f32_16x16x4_f32(
                /*neg_a=*/false, a, /*neg_b=*/false, bf[kb],
                /*c_mod=*/(short)0, c, /*reuse_a=*/false, /*reuse_b=*/false);
        }

        // C/D layout: VGPR j = row (j, +8 for lanes 16-31), col = lane&15
        const int mhi = (lane >> 4) << 3;
#pragma unroll
        for (int j = 0; j < 8; ++j) {
            int r = rowbase + j + mhi;
            if (r < nrows) Y[(size_t)r * DF + colbase + m] = c[j];
        }
        __syncthreads();  // all waves done reading As[buf] before its reuse
    }
}

// agg[i,:] = bias + y[i,:] * dinv[i]^2   (bias + fused self-loop), float4-wide
__global__ void init_agg_v4(const float4* __restrict__ y4,
                            const float* __restrict__ dinv,
                            const float4* __restrict__ bias4,
                            float4* __restrict__ agg4, long long nq) {
    long long i = (long long)blockIdx.x * blockDim.x + threadIdx.x;
    if (i < nq) {
        int node = (int)(i >> 5);        // DF/4 = 32 quads per row
        int q    = (int)(i & 31);
        float di = dinv[node];
        float s  = di * di;
        float4 v = y4[i], b = bias4[q];
        float4 o = {b.x + v.x * s, b.y + v.y * s, b.z + v.z * s, b.w + v.w * s};
        agg4[i] = o;
    }
}

// one wave32 per edge: each lane moves float4 (32*4 = 128 floats = one row);
// edge index forced wave-uniform (scalar loads); atomics are L2-resident
// (51 MB agg << 192 MB global L2)
__global__ void scatter_edges(const float* __restrict__ y, const float* __restrict__ dinv,
                              const int* __restrict__ src, const int* __restrict__ dst,
                              float* __restrict__ agg, int E) {
    long long idx = (long long)blockIdx.x * blockDim.x + threadIdx.x;
    int e = __builtin_amdgcn_readfirstlane((int)(idx >> 5));
    if (e >= E) return;
    int part = (threadIdx.x & 31) << 2;  // 4-float chunk within the row
    int s = src[e], d = dst[e];
    float nrm = dinv[s] * dinv[d];
    float4 v = *(const float4*)(y + (size_t)s * DF + part);
    float* o = agg + (size_t)d * DF + part;
    unsafeAtomicAdd(o + 0, v.x * nrm);
    unsafeAtomicAdd(o + 1, v.y * nrm);
    unsafeAtomicAdd(o + 2, v.z * nrm);
    unsafeAtomicAdd(o + 3, v.w * nrm);
}

__global__ void relu_v4(const float4* __restrict__ in, float4* __restrict__ out,
                        long long nq) {
    long long i = (long long)blockIdx.x * blockDim.x + threadIdx.x;
    if (i < nq) {
        float4 v = in[i];
        float4 o = {fmaxf(v.x, 0.f), fmaxf(v.y, 0.f), fmaxf(v.z, 0.f), fmaxf(v.w, 0.f)};
        out[i] = o;
    }
}

// ---------------------------------------------------------------- launcher
extern "C" void kernel_launch(void* const* d_in, const int* in_sizes, int n_in,
                              void* d_out, int out_size, void* d_ws, size_t ws_size,
                              hipStream_t stream) {
    const float* x  = (const float*)d_in[0];
    const float* W1 = (const float*)d_in[1];
    const float* b1 = (const float*)d_in[2];
    const float* W2 = (const float*)d_in[3];
    const float* b2 = (const float*)d_in[4];
    const int*   ei = (const int*)d_in[5];

    const int N = in_sizes[0] / DF;
    const int E = in_sizes[5] / 2;
    const int* src = ei;
    const int* dst = ei + E;

    // workspace: dinv[N] | bufY[N*DF] | bufH[N*DF]   (all 16 B aligned)
    float* dinv = (float*)d_ws;
    float* bufY = dinv + N;
    float* bufH = bufY + (size_t)N * DF;
    float* out  = (float*)d_out;

    const int T = 256;
    const long long nq = (long long)N * (DF / 4);          // float4 elements
    const int nbN    = (N + T - 1) / T;
    const int nbE    = (E + T - 1) / T;
    const int nbNQ   = (int)((nq + T - 1) / T);
    const long long scat = (long long)E * 32;
    const int nbScat = (int)((scat + T - 1) / T);
    const int nbGemm = (N + 16 * SLABS - 1) / (16 * SLABS);

    // normalization
    zero_f32<<<nbN, T, 0, stream>>>(dinv, N);
    deg_count<<<nbE, T, 0, stream>>>(dst, dinv, E);
    deg_finalize<<<nbN, T, 0, stream>>>(dinv, N);

    // ---- layer 1: h1 = relu(scatter(norm * (x@W1)[src]) + b1)
    gemm_wmma_f32<<<nbGemm, T, 0, stream>>>(x, W1, bufY, N);
    init_agg_v4<<<nbNQ, T, 0, stream>>>((const float4*)bufY, dinv,
                                        (const float4*)b1, (float4*)bufH, nq);
    scatter_edges<<<nbScat, T, 0, stream>>>(bufY, dinv, src, dst, bufH, E);
    relu_v4<<<nbNQ, T, 0, stream>>>((const float4*)bufH, (float4*)bufY, nq);

    // ---- layer 2: out = relu(scatter(norm * (h1@W2)[src]) + b2)
    gemm_wmma_f32<<<nbGemm, T, 0, stream>>>(bufY, W2, bufH, N);
    init_agg_v4<<<nbNQ, T, 0, stream>>>((const float4*)bufH, dinv,
                                        (const float4*)b2, (float4*)out, nq);
    scatter_edges<<<nbScat, T, 0, stream>>>(bufH, dinv, src, dst, out, E);
    relu_v4<<<nbNQ, T, 0, stream>>>((const float4*)out, (float4*)out, nq);
}